// GCN_2680059593310
// MI455X (gfx1250) — compile-verified
//
#include <hip/hip_runtime.h>
#include <math.h>

#define F_IN  264
#define F_HID 48
#define F_CAT 144
#define F_MID 64

typedef float v2f __attribute__((ext_vector_type(2)));
typedef float v8f __attribute__((ext_vector_type(8)));

__device__ __forceinline__ float relu_f(float v) { return v > 0.0f ? v : 0.0f; }

// ---- init: deg = 1 (self-loop weight), outM = -inf (segment_max identity), outA = 0
__global__ void k_init(float* __restrict__ deg, float* __restrict__ oM,
                       float* __restrict__ oA, int N) {
  int t = blockIdx.x * blockDim.x + threadIdx.x;
  int tot = N * F_HID;
  if (t < tot) { oM[t] = -__builtin_inff(); oA[t] = 0.0f; }
  if (t < N) deg[t] = 1.0f;
}

// ---- degree accumulation over col
__global__ void k_deg(float* __restrict__ deg, const int* __restrict__ col,
                      const float* __restrict__ w, int E) {
  int t = blockIdx.x * blockDim.x + threadIdx.x;
  if (t < E) atomicAdd(&deg[col[t]], w[t]);
}

// ---- deg -> d^{-1/2} in place
__global__ void k_dis(float* __restrict__ deg, int N) {
  int t = blockIdx.x * blockDim.x + threadIdx.x;
  if (t < N) { float d = deg[t]; deg[t] = (d > 0.0f) ? rsqrtf(d) : 0.0f; }
}

// ---- per-entry norm: edges then self-loops appended
__global__ void k_norm(float* __restrict__ nrm, const float* __restrict__ dis,
                       const int* __restrict__ row, const int* __restrict__ col,
                       const float* __restrict__ w, int E, int N) {
  int t = blockIdx.x * blockDim.x + threadIdx.x;
  if (t < E) {
    nrm[t] = dis[row[t]] * w[t] * dis[col[t]];
  } else if (t < E + N) {
    int i = t - E;
    float d = dis[i];
    nrm[t] = d * d;   // self-loop weight 1
  }
}

// ---- concat [WM | WA | WS] -> Wcat[264][144]
__global__ void k_cat(float* __restrict__ Wcat, const float* __restrict__ WM,
                      const float* __restrict__ WA, const float* __restrict__ WS) {
  int t = blockIdx.x * blockDim.x + threadIdx.x;
  if (t >= F_IN * F_CAT) return;
  int k = t / F_CAT, j = t - k * F_CAT;
  float v;
  if (j < 48)       v = WM[k * 48 + j];
  else if (j < 96)  v = WA[k * 48 + (j - 48)];
  else              v = WS[k * 48 + (j - 96)];
  Wcat[t] = v;
}

// ---- pack B into V_WMMA_F32_16X16X4_F32 lane layout:
// dst[((ks*ntiles + tile)*32 + lane)*2 + v] = src[(4*ks + 2*(lane/16) + v)*ncol + 16*tile + lane%16]
__global__ void k_pack(float* __restrict__ dst, const float* __restrict__ src,
                       int ncol, int ntiles, int total) {
  int t = blockIdx.x * blockDim.x + threadIdx.x;
  if (t >= total) return;
  int lane = t & 31;
  int tmp  = t >> 5;
  int tile = tmp % ntiles;
  int ks   = tmp / ntiles;
  int n = tile * 16 + (lane & 15);
  int k = ks * 4 + (lane >> 4) * 2;
  dst[(size_t)t * 2 + 0] = src[(size_t)k * ncol + n];
  dst[(size_t)t * 2 + 1] = src[(size_t)(k + 1) * ncol + n];
}

// ---- GEMM1: h[N,144] = x[N,264] @ Wcat ; fp32 WMMA 16x16x4, 64 rows/block, 4 waves
__global__ __launch_bounds__(128) void k_gemm1(const float* __restrict__ x,
                                               const float* __restrict__ Wp,
                                               float* __restrict__ h, int N) {
  extern __shared__ float xt[];           // 64 x 266 (even pitch: 8B-aligned b64, no bank conflicts)
  const int PITCH = 266;
  const int rowbase = blockIdx.x * 64;
  for (int t = threadIdx.x; t < 64 * F_IN; t += 128) {
    int r = t / F_IN, c = t - r * F_IN;
    int gr = rowbase + r;
    xt[r * PITCH + c] = (gr < N) ? x[(size_t)gr * F_IN + c] : 0.0f;
  }
  __syncthreads();

  const int lane = threadIdx.x & 31;
  const int wave = threadIdx.x >> 5;
  const int m  = lane & 15;
  const int kh = lane >> 4;
  const float* arow = &xt[(wave * 16 + m) * PITCH + 2 * kh];

  v8f acc[9] = {};
  const float* bp = Wp + lane * 2;
  for (int ks = 0; ks < 66; ++ks) {
    v2f a = *(const v2f*)(arow + 4 * ks);           // ds_load_b64
    const float* bk = bp + ks * (9 * 64);
    __builtin_prefetch(bk + 9 * 64, 0, 3);          // WGP-scope prefetch of next K-step's B
#pragma unroll
    for (int t = 0; t < 9; ++t) {
      v2f b = *(const v2f*)(bk + t * 64);           // coalesced global_load_b64
      acc[t] = __builtin_amdgcn_wmma_f32_16x16x4_f32(
          false, a, false, b, (short)0, acc[t], false, false);
    }
  }
#pragma unroll
  for (int t = 0; t < 9; ++t) {
#pragma unroll
    for (int r = 0; r < 8; ++r) {
      int row = rowbase + wave * 16 + 8 * kh + r;   // D: M = vgpr + 8*(lane/16)
      if (row < N) h[(size_t)row * F_CAT + t * 16 + m] = acc[t][r];
    }
  }
}

// ---- edge scatter: msg = norm * h[row]; scatter-max (native f32 max atomic) + scatter-add
__global__ void k_edge(const float* __restrict__ h, const float* __restrict__ nrm,
                       const int* __restrict__ row, const int* __restrict__ col,
                       float* __restrict__ oM, float* __restrict__ oA, int E, int N) {
  unsigned t = blockIdx.x * blockDim.x + threadIdx.x;
  unsigned tot = (unsigned)(E + N) * 48u;
  if (t >= tot) return;
  int e = (int)(t / 48u);
  int j = (int)(t - (unsigned)e * 48u);
  int r, c;
  if (e < E) { r = row[e]; c = col[e]; }
  else       { r = e - E; c = r; }
  float nv = nrm[e];
  const float* hr = h + (size_t)r * F_CAT;
  float mM = nv * hr[j];
  float mA = nv * hr[48 + j];
  float* pM = oM + (size_t)c * 48 + j;
  asm volatile("global_atomic_max_num_f32 %0, %1, off"
               :: "v"(pM), "v"(mM) : "memory");
  atomicAdd(oA + (size_t)c * 48 + j, mA);
}

// ---- GEMM2: hmid[N,64] = relu( relu-concat[N,144] @ W1 + b1 )
__global__ __launch_bounds__(128) void k_gemm2(
    const float* __restrict__ oM, const float* __restrict__ oA,
    const float* __restrict__ h,  const float* __restrict__ bM,
    const float* __restrict__ bA, const float* __restrict__ bS,
    const float* __restrict__ Wp, const float* __restrict__ b1,
    float* __restrict__ hm, int N) {
  extern __shared__ float ct[];           // 64 x 146
  const int PITCH = 146;
  const int rowbase = blockIdx.x * 64;
  for (int t = threadIdx.x; t < 64 * F_CAT; t += 128) {
    int r = t / F_CAT, j = t - r * F_CAT;
    int gr = rowbase + r;
    float v = 0.0f;
    if (gr < N) {
      if (j < 48)      v = relu_f(oM[(size_t)gr * 48 + j] + bM[j]);
      else if (j < 96) v = relu_f(oA[(size_t)gr * 48 + (j - 48)] + bA[j - 48]);
      else             v = relu_f(h[(size_t)gr * F_CAT + j] + bS[j - 96]);
    }
    ct[r * PITCH + j] = v;
  }
  __syncthreads();

  const int lane = threadIdx.x & 31;
  const int wave = threadIdx.x >> 5;
  const int m  = lane & 15;
  const int kh = lane >> 4;
  const float* arow = &ct[(wave * 16 + m) * PITCH + 2 * kh];

  v8f acc[4] = {};
  const float* bp = Wp + lane * 2;
  for (int ks = 0; ks < 36; ++ks) {
    v2f a = *(const v2f*)(arow + 4 * ks);
    const float* bk = bp + ks * (4 * 64);
    __builtin_prefetch(bk + 4 * 64, 0, 3);          // WGP-scope prefetch of next K-step's B
#pragma unroll
    for (int t = 0; t < 4; ++t) {
      v2f b = *(const v2f*)(bk + t * 64);
      acc[t] = __builtin_amdgcn_wmma_f32_16x16x4_f32(
          false, a, false, b, (short)0, acc[t], false, false);
    }
  }
#pragma unroll
  for (int t = 0; t < 4; ++t) {
    float bb = b1[t * 16 + m];
#pragma unroll
    for (int r = 0; r < 8; ++r) {
      int row = rowbase + wave * 16 + 8 * kh + r;
      if (row < N) hm[(size_t)row * F_MID + t * 16 + m] = relu_f(acc[t][r] + bb);
    }
  }
}

// ---- final matvec: out[i] = hmid[i,:] . W2 + b2
__global__ void k_out(const float* __restrict__ hm, const float* __restrict__ W2,
                      const float* __restrict__ b2, float* __restrict__ out, int N) {
  int i = blockIdx.x * blockDim.x + threadIdx.x;
  if (i >= N) return;
  const float* hp = hm + (size_t)i * F_MID;
  float s = b2[0];
#pragma unroll
  for (int k = 0; k < F_MID; ++k) s = fmaf(hp[k], W2[k], s);
  out[i] = s;
}

extern "C" void kernel_launch(void* const* d_in, const int* in_sizes, int n_in,
                              void* d_out, int out_size, void* d_ws, size_t ws_size,
                              hipStream_t stream) {
  const float* x  = (const float*)d_in[0];
  const int*   ei = (const int*)d_in[1];
  const float* ew = (const float*)d_in[2];
  const float* WM = (const float*)d_in[3];
  const float* bM = (const float*)d_in[4];
  const float* WA = (const float*)d_in[5];
  const float* bA = (const float*)d_in[6];
  const float* WS = (const float*)d_in[7];
  const float* bS = (const float*)d_in[8];
  const float* W1 = (const float*)d_in[9];
  const float* b1 = (const float*)d_in[10];
  const float* W2 = (const float*)d_in[11];
  const float* b2 = (const float*)d_in[12];
  float* out = (float*)d_out;

  const int N = in_sizes[0] / F_IN;
  const int E = in_sizes[1] / 2;
  const int* row = ei;
  const int* col = ei + E;

  // workspace partition (floats)
  float* ws   = (float*)d_ws;
  float* deg  = ws;                               // N  (becomes d^{-1/2})
  float* nrm  = deg + N;                          // E+N
  float* h    = nrm + (size_t)(E + N);            // N*144
  float* oM   = h  + (size_t)N * F_CAT;           // N*48
  float* oA   = oM + (size_t)N * F_HID;           // N*48
  float* hm   = oA + (size_t)N * F_HID;           // N*64
  float* Wcat = hm + (size_t)N * F_MID;           // 264*144
  float* Wp1  = Wcat + F_IN * F_CAT;              // 66*9*64
  float* Wp2  = Wp1 + 66 * 9 * 64;                // 36*4*64

  const int B = 256;
  k_init<<<(N * F_HID + B - 1) / B, B, 0, stream>>>(deg, oM, oA, N);
  k_deg <<<(E + B - 1) / B, B, 0, stream>>>(deg, col, ew, E);
  k_dis <<<(N + B - 1) / B, B, 0, stream>>>(deg, N);
  k_norm<<<(E + N + B - 1) / B, B, 0, stream>>>(nrm, deg, row, col, ew, E, N);
  k_cat <<<(F_IN * F_CAT + B - 1) / B, B, 0, stream>>>(Wcat, WM, WA, WS);
  k_pack<<<(66 * 9 * 32 + B - 1) / B, B, 0, stream>>>(Wp1, Wcat, F_CAT, 9, 66 * 9 * 32);
  k_pack<<<(36 * 4 * 32 + B - 1) / B, B, 0, stream>>>(Wp2, W1, F_MID, 4, 36 * 4 * 32);

  const int g1blocks = (N + 63) / 64;
  k_gemm1<<<g1blocks, 128, 64 * 266 * sizeof(float), stream>>>(x, Wp1, h, N);

  unsigned etot = (unsigned)(E + N) * 48u;
  k_edge<<<(etot + B - 1) / B, B, 0, stream>>>(h, nrm, row, col, oM, oA, E, N);

  k_gemm2<<<g1blocks, 128, 64 * 146 * sizeof(float), stream>>>(
      oM, oA, h, bM, bA, bS, Wp2, b1, hm, N);

  k_out<<<(N + B - 1) / B, B, 0, stream>>>(hm, W2, b2, out, N);
}